// RoiAlign_8358006358565
// MI455X (gfx1250) — compile-verified
//
#include <hip/hip_runtime.h>
#include <hip/hip_bf16.h>

// ROI-align on MI455X (gfx1250, wave32) via separable interpolation cast as
// two chained 16x16x16 f32 GEMMs per (box, channel):
//     Out(7x7) = A(7x14) * P(14x14) * B^T(14x7)   (padded to 16x16)
// executed with 8x V_WMMA_F32_16X16X4_F32 per wave. The op is gather/bandwidth
// bound (~few hundred MB of cacheline traffic @ 23.3 TB/s ~= 15-25us floor);
// WMMA does the interpolation arithmetic for free on the tensor pipe while the
// memory path stays the minimal 8-gathers-per-lane pattern.

typedef __attribute__((ext_vector_type(2))) float v2f;
typedef __attribute__((ext_vector_type(8))) float v8f;

#define FM_B 4
#define FM_C 256
#define FM_H 200
#define FM_W 304
#define NBOX 1024
#define ES   7

// cc = min(i*step + start, stop); returns floor idx, ceil idx, fractional part.
__device__ __forceinline__ void coord1(float start, float stop, float step, int i,
                                       int& i0, int& i1, float& fr) {
  float cc = fminf(fmaf((float)i, step, start), stop);
  float fl = floorf(cc);
  i0 = (int)fl;
  i1 = (int)ceilf(cc);
  fr = cc - fl;
}

// After GEMM1 the 16x16 product N1 sits in C/D layout: VGPR v holds
// N1[v + 8*h][m] (m = lane&15, h = lane>>4). ROWV(r) fetches N1[r][m] for any
// r in [0,16) using the local regs plus the half-wave-swapped copies.
#define ROWV(r) (((r) < 8) ? (h ? swp[(r)] : loc[(r)]) : (h ? loc[(r)-8] : swp[(r)-8]))

__global__ __launch_bounds__(256) void roi_align_wmma(
    const float* __restrict__ fm, const float* __restrict__ boxes,
    const int* __restrict__ assoc, float* __restrict__ out) {
  const int n    = blockIdx.y;                 // box
  const int c    = blockIdx.x * 8 + (threadIdx.x >> 5);  // channel (1 per wave)
  const int lane = threadIdx.x & 31;
  const int m    = lane & 15;
  const int h    = lane >> 4;

  // Box geometry: boxes is (N,2,2) = [start(y,x), stop(y,x)].
  const float sy = boxes[n * 4 + 0], sx = boxes[n * 4 + 1];
  const float py = boxes[n * 4 + 2], px = boxes[n * 4 + 3];
  const float stepy = (py - sy) * (1.0f / (float)(ES - 1));
  const float stepx = (px - sx) * (1.0f / (float)(ES - 1));
  const int b = assoc[n];

  // Patch row for this lane: y-slot m = 2*iy + (floor/ceil selector).
  int y0, y1;
  float fry;
  coord1(sy, py, stepy, m >> 1, y0, y1, fry);
  const int  yrow = (m & 1) ? y1 : y0;
  const bool vrow = (m < 14);

  // Weight fractions for weight-matrix rows/cols i = m (valid for m < 7).
  int   td0, td1;
  float fym, fxm;
  coord1(sy, py, stepy, m, td0, td1, fym);
  coord1(sx, px, stepx, m, td0, td1, fxm);

  const float* plane = fm + ((size_t)(b * FM_C + c)) * (size_t)(FM_H * FM_W);
  const size_t roff  = (size_t)yrow * FM_W;

  // GEMM1: N1 = P(16x16) x Bt(16x16).
  //   P as A-operand: lane supplies P[m][4q+2h], P[m][4q+2h+1]  (8 gathers).
  //   Bt as B-operand: Bt[s][j], lane's j = m; nonzero iff s == 2j or 2j+1.
  v2f pa[4], bt[4];
#pragma unroll
  for (int q = 0; q < 4; ++q) {
    const int ix = 2 * q + h;  // x-slot pair index (columns 2*ix, 2*ix+1)
    int   x0, x1;
    float frx;
    coord1(sx, px, stepx, ix, x0, x1, frx);
    const bool vcol = (ix < 7);
    pa[q].x = (vrow && vcol) ? plane[roff + x0] : 0.0f;
    pa[q].y = (vrow && vcol) ? plane[roff + x1] : 0.0f;
    const bool sel = (2 * q + h == m) && (m < 7);
    bt[q].x = sel ? (1.0f - fxm) : 0.0f;
    bt[q].y = sel ? fxm : 0.0f;
  }

  v8f n1 = {0.f, 0.f, 0.f, 0.f, 0.f, 0.f, 0.f, 0.f};
#pragma unroll
  for (int q = 0; q < 4; ++q)
    n1 = __builtin_amdgcn_wmma_f32_16x16x4_f32(false, pa[q], false, bt[q],
                                               (short)0, n1, false, false);

  // Re-layout N1 (D layout) into B-operand chunks: lane needs N1[4q+2h+e][m].
  float loc[8], swp[8];
#pragma unroll
  for (int v = 0; v < 8; ++v) {
    loc[v] = n1[v];
    swp[v] = __shfl_xor(n1[v], 16, 32);  // half-wave swap (wave32)
  }

  // GEMM2: Out = A(16x16) x N1(16x16).
  //   A as A-operand: A[m][s], nonzero iff s == 2m or 2m+1 (m < 7).
  v8f acc = {0.f, 0.f, 0.f, 0.f, 0.f, 0.f, 0.f, 0.f};
#pragma unroll
  for (int q = 0; q < 4; ++q) {
    v2f aq, bq;
    const bool sel = (2 * q + h == m) && (m < 7);
    aq.x = sel ? (1.0f - fym) : 0.0f;
    aq.y = sel ? fym : 0.0f;
    bq.x = h ? ROWV(4 * q + 2) : ROWV(4 * q + 0);
    bq.y = h ? ROWV(4 * q + 3) : ROWV(4 * q + 1);
    acc = __builtin_amdgcn_wmma_f32_16x16x4_f32(false, aq, false, bq,
                                                (short)0, acc, false, false);
  }

  // D layout: VGPR v, lane m (h==0) holds Out[v][m]; valid region 7x7.
  if (h == 0 && m < 7) {
    float* o = out + (((size_t)n * FM_C + c) * ES) * ES;
#pragma unroll
    for (int v = 0; v < 7; ++v) o[v * ES + m] = acc[v];
  }
}

extern "C" void kernel_launch(void* const* d_in, const int* in_sizes, int n_in,
                              void* d_out, int out_size, void* d_ws, size_t ws_size,
                              hipStream_t stream) {
  const float* fm    = (const float*)d_in[0];
  const float* boxes = (const float*)d_in[1];
  const int*   assoc = (const int*)d_in[2];
  float*       out   = (float*)d_out;
  (void)in_sizes; (void)n_in; (void)out_size; (void)d_ws; (void)ws_size;

  dim3 grid(FM_C / 8, NBOX);  // 8 waves/block -> 8 channels of one box
  roi_align_wmma<<<grid, dim3(256), 0, stream>>>(fm, boxes, assoc, out);
}